// SpatialTransformNetwork_24111946400521
// MI455X (gfx1250) — compile-verified
//
#include <hip/hip_runtime.h>

// ---------------------------------------------------------------------------
// SpatialTransformNetwork (3D trilinear warp, zeros padding, align_corners).
//   out[b,0,d,h,w] = sum_{8 taps} src[b,0,clamp(...)] * wd*wh*ww * inbounds
// HBM-bound (~236 MB moved, ~10us roofline @ 23.3 TB/s). No matrix structure:
// WMMA intentionally NOT used (per-lane data-dependent gather). grid input is
// recomputed from indices (saves 118 MB of reads). Flow/out use CDNA5 NT
// temporal hints; src stays RT so the 39 MB volume lives in the 192 MB L2.
// Launch geometry carries (b,d) in blockIdx.z/y so per-thread index math is a
// single divide-by-80 instead of a 4-level magic-multiply chain.
// ---------------------------------------------------------------------------

typedef float v2f __attribute__((ext_vector_type(2)));

constexpr int Dd = 160, Hh = 192, Ww = 160, Bb = 2;
constexpr int HW  = Hh * Ww;               // 30720
constexpr int DHW = Dd * HW;               // 4,915,200
constexpr int W2  = Ww / 2;                // 80 pairs per row
constexpr int PAIRS_PER_SLICE = Hh * W2;   // 15360 = 60 * 256 exactly
constexpr int BLOCKS_PER_SLICE = PAIRS_PER_SLICE / 256;  // 60

// Reference does: n = 2*(c/(S-1) - 0.5); p = (n+1)*0.5*(S-1).  Identity up to
// rounding; reproduce the FP ops with a compile-time reciprocal.
__device__ __forceinline__ float norm_roundtrip(float c, float Sm1, float rSm1) {
    float n = 2.0f * (c * rSm1 - 0.5f);
    return (n + 1.0f) * 0.5f * Sm1;
}

__device__ __forceinline__ float sample_one(const float* __restrict__ s,
                                            float cd, float ch, float cw) {
    float d0f = floorf(cd), h0f = floorf(ch), w0f = floorf(cw);
    float fd = cd - d0f,    fh = ch - h0f,    fw = cw - w0f;
    float gd = 1.0f - fd,   gh = 1.0f - fh,   gw = 1.0f - fw;
    int d0 = (int)d0f, h0 = (int)h0f, w0 = (int)w0f;

    // Fast path: fully interior cube -> 8 unmasked taps, src loads hit L0/L2.
    if (d0 >= 0 && d0 < Dd - 1 && h0 >= 0 && h0 < Hh - 1 &&
        w0 >= 0 && w0 < Ww - 1) {
        const float* p = s + (d0 * HW + h0 * Ww + w0);
        float r;
        r = p[0]          * (gd * gh * gw);
        r = fmaf(p[1],          gd * gh * fw, r);
        r = fmaf(p[Ww],         gd * fh * gw, r);
        r = fmaf(p[Ww + 1],     gd * fh * fw, r);
        r = fmaf(p[HW],         fd * gh * gw, r);
        r = fmaf(p[HW + 1],     fd * gh * fw, r);
        r = fmaf(p[HW + Ww],    fd * fh * gw, r);
        r = fmaf(p[HW + Ww + 1],fd * fh * fw, r);
        return r;
    }

    // Border / out-of-bounds path: clamp indices, zero the weight when OOB.
    float r = 0.0f;
    float wdv[2] = {gd, fd}, whv[2] = {gh, fh}, wwv[2] = {gw, fw};
    #pragma unroll
    for (int dd = 0; dd < 2; ++dd) {
        int di = d0 + dd;
        #pragma unroll
        for (int hh2 = 0; hh2 < 2; ++hh2) {
            int hi = h0 + hh2;
            #pragma unroll
            for (int ww2 = 0; ww2 < 2; ++ww2) {
                int wi = w0 + ww2;
                bool inb = ((unsigned)di < (unsigned)Dd) &
                           ((unsigned)hi < (unsigned)Hh) &
                           ((unsigned)wi < (unsigned)Ww);
                int dic = min(max(di, 0), Dd - 1);
                int hic = min(max(hi, 0), Hh - 1);
                int wic = min(max(wi, 0), Ww - 1);
                float v = s[dic * HW + hic * Ww + wic];
                float wgt = inb ? (wdv[dd] * whv[hh2] * wwv[ww2]) : 0.0f;
                r = fmaf(v, wgt, r);
            }
        }
    }
    return r;
}

__global__ __launch_bounds__(256)
void stn_warp_kernel(const float* __restrict__ src,
                     const float* __restrict__ flow,
                     float* __restrict__ out) {
    // (b, d) ride in blockIdx.z / blockIdx.y (SALU-side, wave-uniform).
    const int b = blockIdx.z;
    const int d = blockIdx.y;

    // In-slice pair index: exact 60 blocks * 256 threads = 15360 pairs.
    int t  = blockIdx.x * 256 + threadIdx.x;
    int h  = t / W2;            // single magic-divide by 80
    int w2 = t - h * W2;

    int pos = (d * Hh + h) * Ww + 2 * w2;          // even -> 8B aligned pairs
    const float* fb = flow + (size_t)b * 3 * DHW;

    // Flow is streamed exactly once: non-temporal b64 loads (TH=NT).
    v2f f0 = __builtin_nontemporal_load((const v2f*)(fb + pos));            // d-comp
    v2f f1 = __builtin_nontemporal_load((const v2f*)(fb + DHW + pos));      // h-comp
    v2f f2 = __builtin_nontemporal_load((const v2f*)(fb + 2 * DHW + pos));  // w-comp

    constexpr float Dm1 = (float)(Dd - 1), Hm1 = (float)(Hh - 1), Wm1 = (float)(Ww - 1);
    constexpr float rD = 1.0f / Dm1, rH = 1.0f / Hm1, rW = 1.0f / Wm1;
    float wf = (float)(2 * w2);

    const float* sb = src + (size_t)b * DHW;

    float cd0 = norm_roundtrip((float)d + f0.x, Dm1, rD);
    float ch0 = norm_roundtrip((float)h + f1.x, Hm1, rH);
    float cw0 = norm_roundtrip(wf        + f2.x, Wm1, rW);
    float cd1 = norm_roundtrip((float)d + f0.y, Dm1, rD);
    float ch1 = norm_roundtrip((float)h + f1.y, Hm1, rH);
    float cw1 = norm_roundtrip(wf + 1.0f + f2.y, Wm1, rW);

    v2f r;
    r.x = sample_one(sb, cd0, ch0, cw0);
    r.y = sample_one(sb, cd1, ch1, cw1);

    // Output written once: non-temporal b64 store.
    __builtin_nontemporal_store(r, (v2f*)(out + (size_t)b * DHW + pos));
}

extern "C" void kernel_launch(void* const* d_in, const int* in_sizes, int n_in,
                              void* d_out, int out_size, void* d_ws, size_t ws_size,
                              hipStream_t stream) {
    const float* src  = (const float*)d_in[0];   // (2,1,160,192,160) f32
    const float* flow = (const float*)d_in[1];   // (2,3,160,192,160) f32
    // d_in[2] is the meshgrid; recomputed on the fly, not loaded.
    float* out = (float*)d_out;                  // (2,1,160,192,160) f32

    (void)in_sizes; (void)n_in; (void)out_size; (void)d_ws; (void)ws_size;

    dim3 grid(BLOCKS_PER_SLICE, Dd, Bb);         // (60, 160, 2)
    stn_warp_kernel<<<grid, 256, 0, stream>>>(src, flow, out);
}